// RNNcell_38388417692501
// MI455X (gfx1250) — compile-verified
//
#include <hip/hip_runtime.h>
#include <math.h>
#include <stdint.h>

// Elman RNN on MI455X (gfx1250), fp32 via native V_WMMA_F32_16X16X4_F32.
//   Phase 1 (parallel):  P[t,b,h] = X[t,b,:]·Wi[h,:] + bi[h] + bh[h]   (one big GEMM)
//   Phase 2 (sequential): H_t = tanh(P_t + H_{t-1}·Wh^T), 512 stream-ordered launches,
//       with the block's Wh tile staged into LDS by the Tensor Data Mover (TDM).
// P / H live in-place in d_out[0 .. S*B*H); h_final copied D2D at the end.

#define SEQ   512
#define BATCH 64
#define EMB   1024
#define HID   1024

typedef __attribute__((ext_vector_type(2))) float        v2f;
typedef __attribute__((ext_vector_type(8))) float        v8f;
typedef __attribute__((ext_vector_type(4))) unsigned int v4u;
typedef __attribute__((ext_vector_type(8))) int          v8i;
typedef __attribute__((ext_vector_type(4))) int          v4i;

// D = A(16x4, f32) * B(4x16, f32) + C   -- full-precision fp32 WMMA
__device__ __forceinline__ v8f wmma_f32_k4(v2f a, v2f b, v8f c) {
  // (neg_a, A, neg_b, B, c_mod, C, reuse_a, reuse_b)
  return __builtin_amdgcn_wmma_f32_16x16x4_f32(false, a, false, b, (short)0, c,
                                               false, false);
}

// ---------------------------------------------------------------------------
// Phase 1: P = X @ Wi^T + (bi + bh).
// X: [SEQ*BATCH, EMB] row-major, Wi: [HID, EMB] row-major (so B-matrix = Wi^T).
// Each wave computes a 64(M) x 64(N) macro-tile = 4x4 grid of 16x16 WMMA tiles.
// Block = 128 threads = 4 waves spread across N. Grid = (HID/256, M/64).
// ---------------------------------------------------------------------------
__global__ __launch_bounds__(128) void rnn_phase1_gemm(
    const float* __restrict__ X,
    const float* __restrict__ Wi,
    const float* __restrict__ bi,
    const float* __restrict__ bh,
    float* __restrict__ P) {
  const int lane = threadIdx.x & 31;
  const int wave = threadIdx.x >> 5;
  const int lm   = lane & 15;   // row/col within fragment
  const int half = lane >> 4;   // K-pair select (ISA 16x4 f32 A/B layout)

  const int row0 = blockIdx.y * 64;                // M base
  const int col0 = blockIdx.x * 256 + wave * 64;   // N base

  v8f acc[4][4] = {};

  const float* aBase = X  + (size_t)(row0 + lm) * EMB + 2 * half;
  const float* bBase = Wi + (size_t)(col0 + lm) * EMB + 2 * half;

  for (int kk = 0; kk < EMB; kk += 4) {
    v2f a[4], b[4];
#pragma unroll
    for (int i = 0; i < 4; ++i)
      a[i] = *(const v2f*)(aBase + (size_t)(i * 16) * EMB + kk);
#pragma unroll
    for (int j = 0; j < 4; ++j)
      b[j] = *(const v2f*)(bBase + (size_t)(j * 16) * EMB + kk);
#pragma unroll
    for (int i = 0; i < 4; ++i)
#pragma unroll
      for (int j = 0; j < 4; ++j)
        acc[i][j] = wmma_f32_k4(a[i], b[j], acc[i][j]);
  }

  // C/D layout: lane half h, VGPR r -> (M = 8*h + r, N = lm). Bias depends on N only.
#pragma unroll
  for (int j = 0; j < 4; ++j) {
    const int n = col0 + j * 16 + lm;
    const float bias = bi[n] + bh[n];
#pragma unroll
    for (int i = 0; i < 4; ++i) {
#pragma unroll
      for (int r = 0; r < 8; ++r) {
        const int m = row0 + i * 16 + 8 * half + r;
        P[(size_t)m * HID + n] = acc[i][j][r] + bias;
      }
    }
  }
}

// ---------------------------------------------------------------------------
// Phase 2 step: PH (in: P_t) -> H_t = tanh(P_t + Hprev @ Wh^T), in place.
// Grid = HID/16 blocks of 128 threads; wave w owns tile (M = 16*w, N = 16*bid).
// Wave 0 issues one TDM tensor_load_to_lds for the 16x1024 f32 Wh tile the
// whole block shares (64 KB of the 320 KB WGP LDS), waits TENSORcnt==0, then
// the barrier releases all 4 waves to read B-fragments from LDS (ds_load_b64).
// 4 interleaved K-partial accumulators break the WMMA accumulate chain.
// ---------------------------------------------------------------------------
__global__ __launch_bounds__(128) void rnn_step(
    const float* __restrict__ Hprev,   // [BATCH, HID] or nullptr (t==0, h0=0)
    const float* __restrict__ Wh,      // [HID, HID] row-major
    float* __restrict__ PH) {          // [BATCH, HID] in/out
  __shared__ float sWh[16 * HID];      // 64 KB: Wh rows [n0, n0+16)

  const int lane = threadIdx.x & 31;
  const int wave = threadIdx.x >> 5;
  const int lm   = lane & 15;
  const int half = lane >> 4;

  const int m0 = wave * 16;            // batch tile
  const int n0 = blockIdx.x * 16;      // hidden tile

  v8f acc = {};
  if (Hprev != nullptr) {              // block-uniform branch (kernarg)
    if (wave == 0) {
      // --- Tensor DMA descriptor (D#), CDNA5 ISA ---
      const unsigned ldsAddr = (unsigned)(uintptr_t)(&sWh[0]);  // low 32b = LDS offset
      const unsigned long long ga =
          (unsigned long long)(uintptr_t)(Wh + (size_t)n0 * HID);
      v4u g0 = {
          1u,                                         // count=1, user descriptor
          ldsAddr,                                    // lds_addr (bytes)
          (unsigned)(ga & 0xFFFFFFFFu),               // global_addr[31:0]
          (unsigned)((ga >> 32) & 0x01FFFFFFu) | (2u << 30)  // addr[56:32] | type=2
      };
      v8i g1 = {
          0x00020000,                   // workgroup_mask=0, data_size=2 (4 bytes)
          (int)((HID & 0xFFFF) << 16),  // tensor_dim0[15:0]   = 1024
          (int)((HID & 0xFFFF) << 16),  // dim0[31:16]=0 | tensor_dim1[15:0] = 1024
          (int)((HID & 0xFFFF) << 16),  // dim1[31:16]=0 | tile_dim0 = 1024
          16,                           // tile_dim1 = 16 rows, tile_dim2 = 0
          HID,                          // tensor_dim0_stride[31:0] = 1024 elements
          0,                            // stride0[47:32]=0, dim1_stride[15:0]=0
          0                             // dim1_stride[47:16]=0
      };
      v4i g2 = {0, 0, 0, 0};            // 2-D tensor: groups 2/3 zero
      v4i g3 = {0, 0, 0, 0};
      v8i g4 = {0, 0, 0, 0, 0, 0, 0, 0};  // extra group (6-arg toolchain), zero
      __builtin_amdgcn_tensor_load_to_lds(g0, g1, g2, g3, g4, 0);
      __builtin_amdgcn_s_wait_tensorcnt(0);
    }
    __syncthreads();

    v8f p0 = {}, p1 = {}, p2 = {}, p3 = {};
    const float* aBase = Hprev + (size_t)(m0 + lm) * HID + 2 * half;
    const float* bBase = &sWh[lm * HID + 2 * half];   // B[k][n] = Wh[n0+n][k] in LDS
    for (int kk = 0; kk < HID; kk += 16) {
      v2f a[4], b[4];
#pragma unroll
      for (int u = 0; u < 4; ++u) {
        a[u] = *(const v2f*)(aBase + kk + 4 * u);
        b[u] = *(const v2f*)(bBase + kk + 4 * u);
      }
      p0 = wmma_f32_k4(a[0], b[0], p0);
      p1 = wmma_f32_k4(a[1], b[1], p1);
      p2 = wmma_f32_k4(a[2], b[2], p2);
      p3 = wmma_f32_k4(a[3], b[3], p3);
    }
    acc = (p0 + p1) + (p2 + p3);
  }

#pragma unroll
  for (int r = 0; r < 8; ++r) {
    const size_t idx = (size_t)(m0 + 8 * half + r) * HID + (n0 + lm);
    PH[idx] = tanhf(PH[idx] + acc[r]);
  }
}

// ---------------------------------------------------------------------------
extern "C" void kernel_launch(void* const* d_in, const int* in_sizes, int n_in,
                              void* d_out, int out_size, void* d_ws, size_t ws_size,
                              hipStream_t stream) {
  const float* x  = (const float*)d_in[0];
  const float* Wi = (const float*)d_in[1];
  const float* bi = (const float*)d_in[2];
  const float* Wh = (const float*)d_in[3];
  const float* bh = (const float*)d_in[4];
  float* out = (float*)d_out;

  // Phase 1: one large GEMM writing P into the output region.
  dim3 g1(HID / 256, (SEQ * BATCH) / 64);
  rnn_phase1_gemm<<<g1, dim3(128), 0, stream>>>(x, Wi, bi, bh, out);

  // Phase 2: 512 sequential steps (stream-ordered, graph-capture safe).
  for (int t = 0; t < SEQ; ++t) {
    const float* hprev = (t == 0) ? nullptr : out + (size_t)(t - 1) * BATCH * HID;
    rnn_step<<<dim3(HID / 16), dim3(128), 0, stream>>>(
        hprev, Wh, out + (size_t)t * BATCH * HID);
  }

  // h_final = H_{S-1}
  (void)hipMemcpyAsync(out + (size_t)SEQ * BATCH * HID,
                       out + (size_t)(SEQ - 1) * BATCH * HID,
                       (size_t)BATCH * HID * sizeof(float),
                       hipMemcpyDeviceToDevice, stream);
}